// CONCATNet_7447473291820
// MI455X (gfx1250) — compile-verified
//
#include <hip/hip_runtime.h>
#include <math.h>

#define Bn 512
#define NWn 256
#define NSn 8
#define Pn 32
#define Dn 512
#define Hn 16
#define An 34

typedef float v2f __attribute__((ext_vector_type(2)));
typedef float v8f __attribute__((ext_vector_type(8)));

struct __align__(16) SM {
  float col[NSn][Dn];     // 16KB  distinct encoded_col rows (stage gather source)
  float wafer[Pn][Dn];    // 64KB  pm_wafer rows (zeros where loc_hold_wafer < 0)
  float pm[Pn][Dn];       // 64KB  pm_emb (WMMA GEMM output)
  float g[2][Dn];         // 4KB   g1, g2 (ll recipes)
  float wdyn[Dn];         // W_pmcat[:,1024:] @ W_dyn[:,0]
  float utk[Dn];          // Wk[:,1024:] @ W_time
  float utv[Dn];          // Wv[:,1024:] @ W_time
  float utsh[Dn];         // Wshk[:,1024:] @ W_time
  float meanv[Dn];
  float ctx[3*Dn];
  float h1[Dn];
  float ctx2[Dn];
  float q[Dn];
  float attout[Dn];
  float mh[Dn];
  float t[2*Dn];          // (mh @ Wshk)[0:1024]
  float z[8][2*Dn];       // 32KB per-wave attention scratch (wvec, then z0|z1)
  float sc[Hn][40];       // per-head scores -> probs
  float fin[40];
  float dscale[40];
  float ninf[40];
  float remain[Pn];
  int   widx[Pn];
  int   stidx[Pn];
  int   rb;
  float ct2;
};

__launch_bounds__(256)
__global__ void fused_forward(
    const float* __restrict__ encoded_row, const float* __restrict__ encoded_col,
    const float* __restrict__ clock_, const float* __restrict__ lpet,
    const int* __restrict__ loc_id, const int* __restrict__ robot_loc,
    const int* __restrict__ loc_hold_wafer, const int* __restrict__ loc_stage,
    const int* __restrict__ ll1, const int* __restrict__ ll2,
    const int* __restrict__ amask,
    const float* __restrict__ W_dyn, const float* __restrict__ W_pmcat,
    const float* __restrict__ W_time, const float* __restrict__ W_llctx,
    const float* __restrict__ W_pmctx, const float* __restrict__ W_rbctx,
    const float* __restrict__ W_cc1, const float* __restrict__ b_cc1,
    const float* __restrict__ W_cc2, const float* __restrict__ b_cc2,
    const float* __restrict__ Wq, const float* __restrict__ Wk,
    const float* __restrict__ Wv, const float* __restrict__ Wshk,
    const float* __restrict__ W_mhc, const float* __restrict__ b_mhc,
    float* __restrict__ out)
{
  __shared__ SM sm;
  const int b    = blockIdx.x;
  const int tid  = threadIdx.x;
  const int lane = tid & 31;
  const int wave = tid >> 5;
  const float clk = clock_[b];
  const int rb    = robot_loc[b];

  // ---------------- phase 0: scalars, indices, masks ----------------
  if (tid < Pn) {
    const int p = tid;
    const float endt = lpet[b*Pn + p];
    sm.remain[p] = fmaxf(endt - clk, 0.f) * (1.f/300.f);
    const int lid = loc_id[b*Pn + p];
    const float pk = clk + 3.f * ((lid != rb) ? 1.f : 0.f);
    sm.dscale[2+p] = (fmaxf(endt, pk) + 9.f - clk) * (1.f/300.f);
    const int hw = loc_hold_wafer[b*Pn + p];
    sm.widx[p]  = (hw >= 0) ? hw : NWn;
    sm.stidx[p] = loc_stage[b*Pn + p] - 1;
  }
  if (tid == 0) {
    const float d0 = (fmaxf(0.f, clk + 3.f*((rb != 0) ? 1.f : 0.f)) + 9.f - clk) * (1.f/300.f);
    sm.dscale[0] = d0; sm.dscale[1] = d0;
    sm.rb = rb;
  }
  if (tid < An) sm.ninf[tid] = amask[b*An + tid] ? 0.f : -1e10f;
  __syncthreads();

  // ---------------- phase 1: stage activations + derived weight vectors ----------------
  { // all 8 encoded_col rows (stage gather source)
    const float4* src = (const float4*)(encoded_col + (size_t)b*NSn*Dn);
    float4* dst = (float4*)&sm.col[0][0];
    for (int i = tid; i < NSn*(Dn/4); i += 256) dst[i] = src[i];
  }
  for (int i = tid; i < Pn*(Dn/4); i += 256) { // pm_wafer rows (zero row for idx==NW)
    const int p = i >> 7, c = i & 127;
    const int wi = sm.widx[p];
    float4 v = make_float4(0.f, 0.f, 0.f, 0.f);
    if (wi < NWn) v = ((const float4*)(encoded_row + ((size_t)b*NWn + wi)*Dn))[c];
    ((float4*)&sm.wafer[p][0])[c] = v;
  }
  {
    const int i1 = ll1[b], i2 = ll2[b];
    for (int c = tid; c < Dn/4; c += 256) {
      ((float4*)&sm.g[0][0])[c] = ((const float4*)(encoded_row + ((size_t)b*NWn + i1)*Dn))[c];
      ((float4*)&sm.g[1][0])[c] = ((const float4*)(encoded_row + ((size_t)b*NWn + i2)*Dn))[c];
    }
  }
  for (int n = tid; n < Dn; n += 256) { // rank-1 factor vectors (K=1024..1535 chunk folded)
    const float* r1 = W_pmcat + (size_t)n*1536 + 1024;
    const float* rk = Wk      + (size_t)n*1536 + 1024;
    const float* rv = Wv      + (size_t)n*1536 + 1024;
    const float* rs = Wshk    + (size_t)n*1536 + 1024;
    float a0=0.f, a1=0.f, a2=0.f, a3=0.f;
    #pragma unroll 4
    for (int j = 0; j < Dn; ++j) {
      const float wt = W_time[j];
      a0 += r1[j]*W_dyn[2*j];
      a1 += rk[j]*wt; a2 += rv[j]*wt; a3 += rs[j]*wt;
    }
    sm.wdyn[n]=a0; sm.utk[n]=a1; sm.utv[n]=a2; sm.utsh[n]=a3;
  }
  __syncthreads();

  // ---------------- phase 2: pm_emb GEMM via V_WMMA_F32_16X16X4_F32 ----------------
  // pm_emb(32x512) = stage(32x512)@Wp0^T + wafer(32x512)@Wp1^T + remain (x) wdyn
  {
    const int mTile = wave & 1;        // 2 M-tiles of 16
    const int nGrp  = wave >> 1;       // 4 groups x 8 N-tiles = 32 N-tiles
    const int l16   = lane & 15;
    const int hi    = lane >> 4;
    const int m     = mTile*16 + l16;
    const float* aRowS = &sm.col[sm.stidx[m]][0];
    const float* aRowW = &sm.wafer[m][0];
    v8f acc[8];
    const v8f vz = {0.f,0.f,0.f,0.f,0.f,0.f,0.f,0.f};
    #pragma unroll
    for (int i = 0; i < 8; ++i) acc[i] = vz;

    #pragma unroll
    for (int chunk = 0; chunk < 2; ++chunk) {
      const float* aPtr = (chunk ? aRowW : aRowS) + 2*hi;
      const float* bBase[8];
      #pragma unroll
      for (int i = 0; i < 8; ++i) {
        const int n = (nGrp*8 + i)*16 + l16;
        bBase[i] = W_pmcat + (size_t)n*1536 + chunk*512 + 2*hi;
      }
      for (int k0 = 0; k0 < 512; k0 += 4) {
        const v2f a = *(const v2f*)(aPtr + k0);       // A[m][k0+2hi .. +1]
        #pragma unroll
        for (int i = 0; i < 8; ++i) {
          const v2f bb = *(const v2f*)(bBase[i] + k0); // B[k][n] = W[n][k]
          acc[i] = __builtin_amdgcn_wmma_f32_16x16x4_f32(
                     false, a, false, bb, (short)0, acc[i], false, false);
        }
      }
    }
    // rank-1 (pm_dyn chunk) + writeback. C layout: VGPR r -> M = r + 8*hi, N = l16.
    #pragma unroll
    for (int i = 0; i < 8; ++i) {
      const int n = (nGrp*8 + i)*16 + l16;
      const float wd = sm.wdyn[n];
      #pragma unroll
      for (int r = 0; r < 8; ++r) {
        const int mr = mTile*16 + hi*8 + r;
        sm.pm[mr][n] = acc[i][r] + sm.remain[mr]*wd;
      }
    }
  }
  __syncthreads();

  // ---------------- phase 3: context chain ----------------
  for (int n = tid; n < Dn; n += 256) {
    float s = 0.f;
    #pragma unroll 4
    for (int p = 0; p < Pn; ++p) s += sm.pm[p][n];
    sm.meanv[n] = s * (1.f/32.f);
  }
  __syncthreads();
  for (int n = tid; n < Dn; n += 256) {
    const float* wl = W_llctx + (size_t)n*1024;
    const float* wp = W_pmctx + (size_t)n*512;
    const float* wr = W_rbctx + (size_t)n*512;
    const float* pr = sm.pm[sm.rb];
    float s0=0.f, s1=0.f, s2=0.f;
    #pragma unroll 4
    for (int j = 0; j < Dn; ++j) {
      s0 += wl[j]*sm.g[0][j] + wl[512+j]*sm.g[1][j];
      s1 += wp[j]*sm.meanv[j];
      s2 += wr[j]*pr[j];
    }
    sm.ctx[n] = s0; sm.ctx[512+n] = s1; sm.ctx[1024+n] = s2;
  }
  __syncthreads();
  for (int n = tid; n < Dn; n += 256) {
    const float* w = W_cc1 + (size_t)n*1536;
    float s = b_cc1[n];
    #pragma unroll 4
    for (int j = 0; j < 1536; ++j) s += w[j]*sm.ctx[j];
    sm.h1[n] = fmaxf(s, 0.f);
  }
  __syncthreads();
  for (int n = tid; n < Dn; n += 256) {
    const float* w = W_cc2 + (size_t)n*512;
    float s = b_cc2[n];
    #pragma unroll 4
    for (int j = 0; j < Dn; ++j) s += w[j]*sm.h1[j];
    sm.ctx2[n] = s;
  }
  __syncthreads();
  for (int n = tid; n < Dn; n += 256) {
    const float* w = Wq + (size_t)n*512;
    float s = 0.f;
    #pragma unroll 4
    for (int j = 0; j < Dn; ++j) s += w[j]*sm.ctx2[j];
    sm.q[n] = s;
  }
  __syncthreads();

  // ---------------- phase 4: head-factored attention (2 heads per wave) ----------------
  for (int hh = 0; hh < 2; ++hh) {
    const int h = wave*2 + hh;
    { // wvec = q_h^T Wk_h  (1024 values -> sm.z[wave])
      float accw[32];
      #pragma unroll
      for (int u = 0; u < 32; ++u) accw[u] = 0.f;
      const float* wkb = Wk + (size_t)(32*h)*1536;
      for (int j = 0; j < 32; ++j) {
        const float qv = sm.q[32*h + j];
        const float* row = wkb + (size_t)j*1536;
        #pragma unroll
        for (int u = 0; u < 16; ++u) accw[u]    += qv*row[u*32 + lane];
        #pragma unroll
        for (int u = 0; u < 16; ++u) accw[16+u] += qv*row[512 + u*32 + lane];
      }
      #pragma unroll
      for (int u = 0; u < 16; ++u) {
        sm.z[wave][u*32 + lane]       = accw[u];
        sm.z[wave][512 + u*32 + lane] = accw[16+u];
      }
    }
    float ukh = 0.f;
    for (int j = 0; j < 32; ++j) ukh += sm.q[32*h + j]*sm.utk[32*h + j];
    __syncthreads();
    // scores over 34 actions (rows 0,1 have zero chunk0; chunk1 = g1/g2)
    for (int a = lane; a < An; a += 32) {
      float s = sm.dscale[a]*ukh;
      if (a >= 2) {
        const float* c0 = sm.pm[a-2];
        const float* c1 = sm.wafer[a-2];
        #pragma unroll 4
        for (int j = 0; j < Dn; ++j) s += sm.z[wave][j]*c0[j] + sm.z[wave][512+j]*c1[j];
      } else {
        const float* c1 = sm.g[a];
        #pragma unroll 4
        for (int j = 0; j < Dn; ++j) s += sm.z[wave][512+j]*c1[j];
      }
      sm.sc[h][a] = s*0.17677669529663687f + sm.ninf[a];
    }
    __syncthreads();
    { // softmax over 34 (within owning wave)
      float mx = -3.4e38f;
      for (int a = 0; a < An; ++a) mx = fmaxf(mx, sm.sc[h][a]);
      float sum = 0.f;
      for (int a = 0; a < An; ++a) sum += expf(sm.sc[h][a] - mx);
      const float inv = 1.f/sum;
      for (int a = lane; a < An; a += 32) sm.sc[h][a] = expf(sm.sc[h][a] - mx)*inv;
    }
    __syncthreads();
    float pd = 0.f;
    for (int a = 0; a < An; ++a) pd += sm.sc[h][a]*sm.dscale[a];
    { // z = sum_a p[a] * action[a] (chunk0|chunk1) -> sm.z[wave]
      float z0[16], z1[16];
      #pragma unroll
      for (int u = 0; u < 16; ++u) { z0[u] = 0.f; z1[u] = 0.f; }
      for (int a = 0; a < 2; ++a) {
        const float pa = sm.sc[h][a];
        #pragma unroll
        for (int u = 0; u < 16; ++u) z1[u] += pa*sm.g[a][u*32 + lane];
      }
      for (int a = 2; a < An; ++a) {
        const float pa = sm.sc[h][a];
        const float* r0 = sm.pm[a-2];
        const float* r1 = sm.wafer[a-2];
        #pragma unroll
        for (int u = 0; u < 16; ++u) { z0[u] += pa*r0[u*32+lane]; z1[u] += pa*r1[u*32+lane]; }
      }
      #pragma unroll
      for (int u = 0; u < 16; ++u) {
        sm.z[wave][u*32 + lane]       = z0[u];
        sm.z[wave][512 + u*32 + lane] = z1[u];
      }
    }
    __syncthreads();
    { // out[32h+lane] = Wv row . z  + pd*utv
      const int r = 32*h + lane;
      const float* wv = Wv + (size_t)r*1536;
      float o = pd*sm.utv[r];
      #pragma unroll 4
      for (int j = 0; j < Dn; ++j) o += wv[j]*sm.z[wave][j] + wv[512+j]*sm.z[wave][512+j];
      sm.attout[r] = o;
    }
    __syncthreads();
  }

  // ---------------- phase 5: mh, t = mh @ Wshk, final scores ----------------
  for (int n = tid; n < Dn; n += 256) {
    const float* w = W_mhc + (size_t)n*512;
    float s = b_mhc[n];
    #pragma unroll 4
    for (int j = 0; j < Dn; ++j) s += w[j]*sm.attout[j];
    sm.mh[n] = s;
  }
  __syncthreads();
  {
    float tacc[4] = {0.f, 0.f, 0.f, 0.f};
    for (int n = 0; n < Dn; ++n) {
      const float m = sm.mh[n];
      const float* row = Wshk + (size_t)n*1536;
      #pragma unroll
      for (int u = 0; u < 4; ++u) tacc[u] += m*row[tid + u*256];
    }
    #pragma unroll
    for (int u = 0; u < 4; ++u) sm.t[tid + u*256] = tacc[u];
  }
  if (tid == 0) {
    float s = 0.f;
    for (int n = 0; n < Dn; ++n) s += sm.mh[n]*sm.utsh[n];
    sm.ct2 = s;
  }
  __syncthreads();
  if (tid < An) {
    const int a = tid;
    float s = sm.dscale[a]*sm.ct2;
    if (a >= 2) {
      const float* c0 = sm.pm[a-2];
      const float* c1 = sm.wafer[a-2];
      #pragma unroll 4
      for (int j = 0; j < Dn; ++j) s += sm.t[j]*c0[j] + sm.t[512+j]*c1[j];
    } else {
      const float* c1 = sm.g[a];
      #pragma unroll 4
      for (int j = 0; j < Dn; ++j) s += sm.t[512+j]*c1[j];
    }
    sm.fin[a] = 10.f*tanhf(s*0.044194173824159216f) + sm.ninf[a];
  }
  __syncthreads();
  if (tid < An) {
    float mx = -3.4e38f;
    for (int a = 0; a < An; ++a) mx = fmaxf(mx, sm.fin[a]);
    float sum = 0.f;
    for (int a = 0; a < An; ++a) sum += expf(sm.fin[a] - mx);
    out[(size_t)b*An + tid] = expf(sm.fin[tid] - mx)/sum;
  }
}

extern "C" void kernel_launch(void* const* d_in, const int* in_sizes, int n_in,
                              void* d_out, int out_size, void* d_ws, size_t ws_size,
                              hipStream_t stream) {
  (void)in_sizes; (void)n_in; (void)out_size; (void)d_ws; (void)ws_size;
  fused_forward<<<Bn, 256, 0, stream>>>(
      (const float*)d_in[0],  (const float*)d_in[1],  (const float*)d_in[2],
      (const float*)d_in[3],  (const int*)d_in[4],    (const int*)d_in[5],
      (const int*)d_in[6],    (const int*)d_in[7],    (const int*)d_in[8],
      (const int*)d_in[9],    (const int*)d_in[10],
      (const float*)d_in[11], (const float*)d_in[12], (const float*)d_in[13],
      (const float*)d_in[14], (const float*)d_in[15], (const float*)d_in[16],
      (const float*)d_in[17], (const float*)d_in[18], (const float*)d_in[19],
      (const float*)d_in[20], (const float*)d_in[21], (const float*)d_in[22],
      (const float*)d_in[23], (const float*)d_in[24], (const float*)d_in[25],
      (const float*)d_in[26], (float*)d_out);
}